// EdgeAtt_58351425683583
// MI455X (gfx1250) — compile-verified
//
#include <hip/hip_runtime.h>

typedef __attribute__((ext_vector_type(2))) float v2f;
typedef __attribute__((ext_vector_type(8))) float v8f;

#define BSZ 64
#define LSZ 512
#define DSZ 512

// Kernel 1: e_i[b,l] = X[b,l,:]·w_a[:D], e_j[b,l] = X[b,l,:]·w_a[D:]
// as a [32768,512] x [512,16] GEMM via V_WMMA_F32_16X16X4_F32.
// One wave per 16-row tile; B tile has w1 in col 0, w2 in col 1, zeros elsewhere.
// Fully unrolled K loop: all loads use immediate offsets from two fixed base
// pointers (2048 B < 24-bit ioffset), 4 independent WMMA accumulator chains.
__global__ __launch_bounds__(256) void edgeatt_gemv_wmma(
    const float* __restrict__ X, const float* __restrict__ w_a,
    float* __restrict__ e_i, float* __restrict__ e_j) {
  const int lane   = threadIdx.x & 31;
  const int waveId = threadIdx.x >> 5;
  const int tile   = blockIdx.x * 8 + waveId;   // 2048 tiles of 16 rows
  const int m0     = tile * 16;
  const int n      = lane & 15;                 // col for B/C, row-within-tile for A
  const int g      = lane >> 4;                 // lane half
  const int row    = m0 + n;                    // A: lanes 0-15 M=0..15, 16-31 M=0..15

  // A layout (16x4 f32): lane half g holds K = 2g, 2g+1 in its two VGPRs.
  const float* arow = X + (size_t)row * DSZ + 2 * g;

  // Branchless B: all lanes load from a valid address; lanes n>=2 are zeroed
  // by a mask multiply (keeps EXEC untouched in the hot loop).
  const int   sel  = (n < 2) ? n : 1;
  const float msk  = (n < 2) ? 1.0f : 0.0f;
  const float* brow = w_a + sel * DSZ + 2 * g;

  v8f c0 = {}, c1 = {}, c2 = {}, c3 = {};
#pragma unroll
  for (int k = 0; k < DSZ; k += 16) {
    // 8 b64 loads with immediate offsets, then 4 WMMAs on 4 separate chains
    v2f a0 = *(const v2f*)(arow + k);
    v2f a1 = *(const v2f*)(arow + k + 4);
    v2f a2 = *(const v2f*)(arow + k + 8);
    v2f a3 = *(const v2f*)(arow + k + 12);
    v2f b0 = *(const v2f*)(brow + k);
    v2f b1 = *(const v2f*)(brow + k + 4);
    v2f b2 = *(const v2f*)(brow + k + 8);
    v2f b3 = *(const v2f*)(brow + k + 12);
    b0 *= msk;
    b1 *= msk;
    b2 *= msk;
    b3 *= msk;
    c0 = __builtin_amdgcn_wmma_f32_16x16x4_f32(
        false, a0, false, b0, (short)0, c0, false, false);
    c1 = __builtin_amdgcn_wmma_f32_16x16x4_f32(
        false, a1, false, b1, (short)0, c1, false, false);
    c2 = __builtin_amdgcn_wmma_f32_16x16x4_f32(
        false, a2, false, b2, (short)0, c2, false, false);
    c3 = __builtin_amdgcn_wmma_f32_16x16x4_f32(
        false, a3, false, b3, (short)0, c3, false, false);
  }
  v8f c = (c0 + c1) + (c2 + c3);

  // C/D layout: lane (n,g), VGPR v holds element (M = v + 8g, N = n).
  if (n < 2) {
    float* dst = (n == 0) ? e_i : e_j;
#pragma unroll
    for (int v = 0; v < 8; ++v)
      dst[m0 + 8 * g + v] = c[v];
  }
}

// Kernel 2: one 256-thread block per output row (b,i); contiguous float2 per thread.
__global__ __launch_bounds__(256) void edgeatt_softmax(
    const float* __restrict__ e_i, const float* __restrict__ e_j,
    const float* __restrict__ b_a, const int* __restrict__ text_len,
    float* __restrict__ A) {
  const int row = blockIdx.x;          // b*L + i
  const int b   = row >> 9;            // L = 512
  const int i   = row & (LSZ - 1);
  const int len = text_len[b];
  float* out = A + (size_t)row * LSZ;
  const int j0 = threadIdx.x * 2;      // this thread's two contiguous columns

  if (i >= len) {                      // uniform per block: row is zeroed
    v2f z = {0.0f, 0.0f};
    *(v2f*)(out + j0) = z;
    return;
  }

  const float  ei  = e_i[row] + b_a[0];
  const float* ejb = e_j + b * LSZ;

  v2f ej = *(const v2f*)(ejb + j0);    // single b64 load

  float x0 = ei + ej.x;
  float x1 = ei + ej.y;
  x0 = (x0 > 0.0f) ? x0 : 0.2f * x0;   // leaky_relu(0.2)
  x1 = (x1 > 0.0f) ? x1 : 0.2f * x1;
  const float v0 = (j0     < len) ? __expf(x0) : 0.0f;  // masked cols -> exactly 0
  const float v1 = (j0 + 1 < len) ? __expf(x1) : 0.0f;

  float s = v0 + v1;

  // wave32 reduction
#pragma unroll
  for (int off = 16; off >= 1; off >>= 1)
    s += __shfl_xor(s, off, 32);

  __shared__ float red[8];
  const int wave = threadIdx.x >> 5;
  if ((threadIdx.x & 31) == 0) red[wave] = s;
  __syncthreads();

  float total = 0.0f;
#pragma unroll
  for (int w = 0; w < 8; ++w) total += red[w];
  const float inv = 1.0f / total;

  v2f res;
  res.x = v0 * inv;
  res.y = v1 * inv;
  *(v2f*)(out + j0) = res;             // single b64 store
}

extern "C" void kernel_launch(void* const* d_in, const int* in_sizes, int n_in,
                              void* d_out, int out_size, void* d_ws, size_t ws_size,
                              hipStream_t stream) {
  const float* X        = (const float*)d_in[0];   // [B, L, D] f32
  const float* w_a      = (const float*)d_in[1];   // [2D] f32
  const float* b_a      = (const float*)d_in[2];   // [1] f32
  const int*   text_len = (const int*)d_in[3];     // [B] int
  float* out = (float*)d_out;                      // [B, L, L] f32

  float* e_i = (float*)d_ws;                       // [B*L] f32
  float* e_j = e_i + BSZ * LSZ;                    // [B*L] f32

  // 2048 tiles / 8 waves per block = 256 blocks
  edgeatt_gemv_wmma<<<(BSZ * LSZ / 16) / 8, 256, 0, stream>>>(X, w_a, e_i, e_j);
  // one block per output row
  edgeatt_softmax<<<BSZ * LSZ, 256, 0, stream>>>(e_i, e_j, b_a, text_len, out);
}